// NeuralConnectionMatrix_81690277970407
// MI455X (gfx1250) — compile-verified
//
#include <hip/hip_runtime.h>

typedef __attribute__((ext_vector_type(2))) float v2f;
typedef __attribute__((ext_vector_type(8))) float v8f;

#define NPTS 4096
#define HD 16

// ---------------------------------------------------------------------------
// Phase 1: L' = fv @ W1[:, :3].T + b1  and  R = fv @ W1[:, 3:].T
// One wave handles 16 rows of fv via V_WMMA_F32_16X16X4_F32 (K=3 padded to 4).
// A layout (16x4 f32): vgpr0 = K0 (lanes 0-15) / K2 (lanes 16-31),
//                      vgpr1 = K1 / K3;  row M = lane%16.
// B layout (4x16 f32): same striping, N = lane%16.
// D layout: vgpr r -> M = r (lanes 0-15) / r+8 (lanes 16-31), N = lane%16.
// ---------------------------------------------------------------------------
__global__ __launch_bounds__(256) void lr_phase_wmma(
    const float* __restrict__ fv, const float* __restrict__ W1,
    const float* __restrict__ b1, float* __restrict__ Rm,
    float* __restrict__ Lp)
{
    const int lane    = threadIdx.x & 31;
    const int wave    = threadIdx.x >> 5;
    const int rowbase = (blockIdx.x * 8 + wave) * 16;
    const int n       = lane & 15;          // M for A, N for B/D
    const bool hi     = lane >= 16;
    const int k0      = hi ? 2 : 0;         // K held in vgpr0
    const int row     = rowbase + n;

    // A: fv tile (16x4), K=3 column zero-padded (selects, no divergent branch)
    float a0 = fv[row * 3 + k0];
    float a1 = hi ? 0.0f : fv[row * 3 + 1];
    v2f A = {a0, a1};

    // B left half: B[k][n] = W1[n][k]   (W1 is 16 x 6 row-major)
    float bl0 = W1[n * 6 + k0];
    float bl1 = hi ? 0.0f : W1[n * 6 + 1];
    v2f BL = {bl0, bl1};
    // B right half: B[k][n] = W1[n][3+k]
    float br0 = W1[n * 6 + 3 + k0];
    float br1 = hi ? 0.0f : W1[n * 6 + 4];
    v2f BR = {br0, br1};

    float bb = b1[n];
    v8f CL = {bb, bb, bb, bb, bb, bb, bb, bb};   // fold b1 into L'
    v8f CR = {0.f, 0.f, 0.f, 0.f, 0.f, 0.f, 0.f, 0.f};

    v8f DL = __builtin_amdgcn_wmma_f32_16x16x4_f32(
        false, A, false, BL, (short)0, CL, false, false);
    v8f DR = __builtin_amdgcn_wmma_f32_16x16x4_f32(
        false, A, false, BR, (short)0, CR, false, false);

    const int mofs = hi ? 8 : 0;
#pragma unroll
    for (int r = 0; r < 8; ++r) {
        const int mm = rowbase + r + mofs;
        Lp[mm * HD + n] = DL[r];
        Rm[mm * HD + n] = DR[r];
    }
}

// ---------------------------------------------------------------------------
// Phase 2: out[i,j] = sum_k w2[k] * relu(R[i,k] + L'[j,k]) + b2
// Block tile: 32 i-rows x 1024 j-cols. Each thread: 4 consecutive j, loop i.
// R tile lives in LDS (broadcast reads); L' rows + w2 live in registers.
// Stores: float4, fully coalesced -> ~2.9us HBM floor for the 67MB output.
// ---------------------------------------------------------------------------
__global__ __launch_bounds__(256) void ncm_main_kernel(
    const float* __restrict__ Rm, const float* __restrict__ Lp,
    const float* __restrict__ W2, const float* __restrict__ b2p,
    float* __restrict__ out)
{
    __shared__ float sR[32 * HD];   // 2 KB
    const int t     = threadIdx.x;
    const int ibase = blockIdx.y * 32;

    // stage R tile: 512 floats, 2 per thread
    {
        const float2* src = (const float2*)(Rm + ibase * HD);
        ((float2*)sR)[t] = src[t];
    }

    const int j0 = blockIdx.x * 1024 + t * 4;

    // L' rows for this thread's 4 j's -> 64 VGPRs (b128 loads)
    float Lc[4][16];
#pragma unroll
    for (int jj = 0; jj < 4; ++jj) {
        const float4* lp = (const float4*)(Lp + (j0 + jj) * HD);
#pragma unroll
        for (int q = 0; q < 4; ++q) {
            float4 v = lp[q];
            Lc[jj][q * 4 + 0] = v.x;
            Lc[jj][q * 4 + 1] = v.y;
            Lc[jj][q * 4 + 2] = v.z;
            Lc[jj][q * 4 + 3] = v.w;
        }
    }

    float w[16];
#pragma unroll
    for (int k = 0; k < 16; ++k) w[k] = W2[k];
    const float b2 = b2p[0];

    __syncthreads();

    for (int ii = 0; ii < 32; ++ii) {
        // wave-uniform R row via LDS broadcast (4x ds_load_b128)
        const float4* rr4 = (const float4*)(sR + ii * HD);
        float rk[16];
#pragma unroll
        for (int q = 0; q < 4; ++q) {
            float4 v = rr4[q];
            rk[q * 4 + 0] = v.x;
            rk[q * 4 + 1] = v.y;
            rk[q * 4 + 2] = v.z;
            rk[q * 4 + 3] = v.w;
        }

        float a0 = b2, a1 = b2, a2 = b2, a3 = b2;
#pragma unroll
        for (int k = 0; k < 16; ++k) {
            float x0 = fmaxf(rk[k] + Lc[0][k], 0.0f);
            float x1 = fmaxf(rk[k] + Lc[1][k], 0.0f);
            float x2 = fmaxf(rk[k] + Lc[2][k], 0.0f);
            float x3 = fmaxf(rk[k] + Lc[3][k], 0.0f);
            a0 = fmaf(x0, w[k], a0);
            a1 = fmaf(x1, w[k], a1);
            a2 = fmaf(x2, w[k], a2);
            a3 = fmaf(x3, w[k], a3);
        }

        float4 res = {a0, a1, a2, a3};
        *((float4*)(out + (size_t)(ibase + ii) * NPTS + j0)) = res;
    }
}

extern "C" void kernel_launch(void* const* d_in, const int* in_sizes, int n_in,
                              void* d_out, int out_size, void* d_ws, size_t ws_size,
                              hipStream_t stream) {
    (void)in_sizes; (void)n_in; (void)out_size; (void)ws_size;
    const float* fv = (const float*)d_in[0];   // (4096, 3)
    const float* W1 = (const float*)d_in[1];   // (16, 6)
    const float* b1 = (const float*)d_in[2];   // (16,)
    const float* W2 = (const float*)d_in[3];   // (1, 16)
    const float* b2 = (const float*)d_in[4];   // (1,)
    float* out = (float*)d_out;                // (4096, 4096) f32

    float* Rm = (float*)d_ws;                  // 4096*16 f32
    float* Lp = Rm + NPTS * HD;                // 4096*16 f32 (total 512 KB)

    // Phase 1: 4096 rows / (16 rows/wave * 8 waves/block) = 32 blocks
    lr_phase_wmma<<<32, 256, 0, stream>>>(fv, W1, b1, Rm, Lp);

    // Phase 2: grid = (4096/1024 j-tiles, 4096/32 i-tiles)
    dim3 grid(4, 128);
    ncm_main_kernel<<<grid, 256, 0, stream>>>(Rm, Lp, W2, b2, out);
}